// LSTMBase_30760555774053
// MI455X (gfx1250) — compile-verified
//
#include <hip/hip_runtime.h>
#include <hip/hip_bf16.h>
#include <math.h>

typedef __attribute__((ext_vector_type(2))) float v2f;
typedef __attribute__((ext_vector_type(8))) float v8f;

namespace {
constexpr int kB = 64;
constexpr int kT = 256;
constexpr int kU = 1024;
constexpr int kV = 256;
constexpr int kG = 4 * kU;   // 4096 gate width
constexpr int kK = 2 * kU;   // 2048 concat K
constexpr int KTILE = 256;   // K chunk staged in LDS
constexpr float kEps = 1e-6f;
constexpr float kForgetBias = 1.0f;
}

__device__ __forceinline__ float sigmoidf_(float x) {
  return 1.0f / (1.0f + __expf(-x));
}

// ---------------------------------------------------------------------------
// z = A(64 x 2048) @ W(2048 x 4096), A = [lo | hi] with lo optionally gathered
// by token index (layer 0 embedding lookup). f32 WMMA 16x16x4.
// grid = (2 mhalves, 32 nblocks), block = 256 (8 waves).
// Each wave computes TWO 16x16 M-tiles for one 16-col stripe, so every B
// fetch (the only per-WMMA global stream) feeds 2 WMMAs: 16 FLOP/B from L2.
// ---------------------------------------------------------------------------
__global__ void __launch_bounds__(256)
lstm_gemm(const float* __restrict__ W,     // (2048, 4096) row-major
          const float* __restrict__ alo,   // cols k <  1024 base
          const float* __restrict__ ahi,   // cols k >= 1024, row-major (64,1024)
          const int*   __restrict__ tok,   // nullptr, or tokens+t (stride T+1)
          int tok_stride,
          float* __restrict__ z)           // (64, 4096)
{
  __shared__ float As[32 * (KTILE + 1)];   // 32 rows, stride 257: no bank conflicts
  __shared__ const float* rowlo[32];
  __shared__ const float* rowhi[32];

  const int mhalf  = blockIdx.x;           // 0..1 -> rows [mhalf*32, mhalf*32+32)
  const int nblock = blockIdx.y;           // 0..31
  const int tid    = threadIdx.x;

  if (tid < 32) {
    const int r = mhalf * 32 + tid;
    rowlo[tid] = tok ? (alo + (size_t)tok[(size_t)r * tok_stride] * kU)
                     : (alo + (size_t)r * kU);
    rowhi[tid] = ahi + (size_t)r * kU;
  }
  __syncthreads();

  const int lane = tid & 31;
  const int wave = tid >> 5;
  const int m    = lane & 15;              // M row (A) / N col (B,C)
  const int kh   = lane >> 4;              // lane-half selector
  const int ncol = nblock * 128 + wave * 16;

  v8f acc0 = {}, acc1 = {};

  for (int k0 = 0; k0 < kK; k0 += KTILE) {
    const bool lo   = (k0 < kU);
    const int kbase = lo ? k0 : (k0 - kU);
    // cooperative A-chunk load: 32 rows x 256 cols, coalesced along K
#pragma unroll
    for (int i = 0; i < 32; ++i) {
      const float* rp = lo ? rowlo[i] : rowhi[i];
      As[i * (KTILE + 1) + tid] = rp[kbase + tid];
    }
    __syncthreads();

    // prefetch next K-chunk of W for this wave's 16-col stripe:
    // 32 lanes x 8 instructions cover all 256 rows (one line each).
    if (k0 + KTILE < kK) {
#pragma unroll
      for (int pk = 0; pk < KTILE; pk += 32) {
        __builtin_prefetch(&W[(size_t)(k0 + KTILE + pk + lane) * kG + ncol], 0, 3);
      }
    }

#pragma unroll 8
    for (int kk = 0; kk < KTILE; kk += 4) {
      const int ksel = kk + 2 * kh;        // VGPR0 holds K=ksel, VGPR1 K=ksel+1
      v2f a0, a1, b;
      a0.x = As[m * (KTILE + 1) + ksel];
      a0.y = As[m * (KTILE + 1) + ksel + 1];
      a1.x = As[(16 + m) * (KTILE + 1) + ksel];
      a1.y = As[(16 + m) * (KTILE + 1) + ksel + 1];
      const size_t krow = (size_t)(k0 + ksel);
      b.x = W[krow * kG + ncol + m];
      b.y = W[(krow + 1) * kG + ncol + m];
      acc0 = __builtin_amdgcn_wmma_f32_16x16x4_f32(false, a0, false, b,
                                                   (short)0, acc0, false, false);
      acc1 = __builtin_amdgcn_wmma_f32_16x16x4_f32(false, a1, false, b,
                                                   (short)0, acc1, false, false);
    }
    __syncthreads();
  }

  // C/D layout: VGPR j -> row j (lanes 0-15) or row j+8 (lanes 16-31), N = lane&15
#pragma unroll
  for (int j = 0; j < 8; ++j) {
    const int row0 = mhalf * 32 + j + 8 * kh;
    const int row1 = row0 + 16;
    z[(size_t)row0 * kG + ncol + m] = acc0[j];
    z[(size_t)row1 * kG + ncol + m] = acc1[j];
  }
}

// ---------------------------------------------------------------------------
// Block reduction (sum, sumsq) over 256 threads, wave32 shuffles + LDS.
// ---------------------------------------------------------------------------
__device__ __forceinline__ void block_reduce2(float& s, float& q, float* sh) {
#pragma unroll
  for (int off = 16; off > 0; off >>= 1) {
    s += __shfl_xor(s, off, 32);
    q += __shfl_xor(q, off, 32);
  }
  const int wave = threadIdx.x >> 5;
  const int lane = threadIdx.x & 31;
  if (lane == 0) { sh[wave] = s; sh[8 + wave] = q; }
  __syncthreads();
  if (threadIdx.x < 32) {
    float s2 = (lane < 8) ? sh[lane] : 0.0f;
    float q2 = (lane < 8) ? sh[8 + lane] : 0.0f;
#pragma unroll
    for (int off = 4; off > 0; off >>= 1) {
      s2 += __shfl_xor(s2, off, 32);
      q2 += __shfl_xor(q2, off, 32);
    }
    if (lane == 0) { sh[0] = s2; sh[8] = q2; }
  }
  __syncthreads();
  s = sh[0];
  q = sh[8];
  __syncthreads();   // safe reuse of sh on next call
}

// ---------------------------------------------------------------------------
// Per-step pointwise: 4 gate LayerNorms + cell update + cell LayerNorm + h.
// grid = 64 (one block per batch row), block = 256 (4 u-elems per thread).
// ---------------------------------------------------------------------------
__global__ void __launch_bounds__(256)
lstm_pointwise(const float* __restrict__ z,      // (64, 4096)
               const float* __restrict__ gamma,  // (5, 1024) for this layer
               const float* __restrict__ beta,   // (5, 1024)
               float* __restrict__ c,            // (64, 1024) in/out
               float* __restrict__ h,            // (64, 1024) out
               float* __restrict__ outs,         // nullptr, or (B,T,U)
               int t)
{
  __shared__ float red[16];
  const int b   = blockIdx.x;
  const int tid = threadIdx.x;
  const float* zb = z + (size_t)b * kG;
  const float inv = 1.0f / (float)kU;

  float gi[4], gj[4], gf[4], go[4];
  float si = 0, qi = 0, sj = 0, qj = 0, sf = 0, qf = 0, so = 0, qo = 0;
#pragma unroll
  for (int k = 0; k < 4; ++k) {
    const int u = tid + k * 256;
    gi[k] = zb[u];
    gj[k] = zb[kU + u];
    gf[k] = zb[2 * kU + u];
    go[k] = zb[3 * kU + u];
    si += gi[k]; qi += gi[k] * gi[k];
    sj += gj[k]; qj += gj[k] * gj[k];
    sf += gf[k]; qf += gf[k] * gf[k];
    so += go[k]; qo += go[k] * go[k];
  }
  block_reduce2(si, qi, red);
  block_reduce2(sj, qj, red);
  block_reduce2(sf, qf, red);
  block_reduce2(so, qo, red);

  const float mu_i = si * inv, r_i = rsqrtf(qi * inv - mu_i * mu_i + kEps);
  const float mu_j = sj * inv, r_j = rsqrtf(qj * inv - mu_j * mu_j + kEps);
  const float mu_f = sf * inv, r_f = rsqrtf(qf * inv - mu_f * mu_f + kEps);
  const float mu_o = so * inv, r_o = rsqrtf(qo * inv - mu_o * mu_o + kEps);

  float nc[4], on[4];
  float sc = 0, qc = 0;
#pragma unroll
  for (int k = 0; k < 4; ++k) {
    const int u = tid + k * 256;
    const float xi = (gi[k] - mu_i) * r_i * gamma[u]          + beta[u];
    const float xj = (gj[k] - mu_j) * r_j * gamma[kU + u]     + beta[kU + u];
    const float xf = (gf[k] - mu_f) * r_f * gamma[2 * kU + u] + beta[2 * kU + u];
    on[k]          = (go[k] - mu_o) * r_o * gamma[3 * kU + u] + beta[3 * kU + u];
    const float cp = c[(size_t)b * kU + u];
    const float v  = cp * sigmoidf_(xf + kForgetBias) + sigmoidf_(xi) * tanhf(xj);
    nc[k] = v;
    sc += v; qc += v * v;
    c[(size_t)b * kU + u] = v;
  }
  block_reduce2(sc, qc, red);
  const float mu_c = sc * inv, r_c = rsqrtf(qc * inv - mu_c * mu_c + kEps);

#pragma unroll
  for (int k = 0; k < 4; ++k) {
    const int u = tid + k * 256;
    const float ln_c = (nc[k] - mu_c) * r_c * gamma[4 * kU + u] + beta[4 * kU + u];
    const float nh = tanhf(ln_c) * sigmoidf_(on[k]);
    h[(size_t)b * kU + u] = nh;
    if (outs) outs[((size_t)b * kT + t) * kU + u] = nh;
  }
}

// ---------------------------------------------------------------------------
// logits = outs(16384 x 1024) @ Wt(1024 x 256) + bias. f32 WMMA.
// grid = 1024 mtiles, block = 256; wave w covers cols [w*32, w*32+32).
// ---------------------------------------------------------------------------
__global__ void __launch_bounds__(256)
logits_gemm(const float* __restrict__ A,     // (16384, 1024)
            const float* __restrict__ Bt,    // (1024, 256) = W_out^T
            const float* __restrict__ bias,  // (256)
            float* __restrict__ out)         // (16384, 256)
{
  __shared__ float As[16 * (KTILE + 1)];
  const int mtile = blockIdx.x;
  const int tid   = threadIdx.x;
  const int lane  = tid & 31;
  const int wave  = tid >> 5;
  const int m     = lane & 15;
  const int kh    = lane >> 4;
  const int n0    = wave * 32;

  v8f acc0 = {}, acc1 = {};

  for (int k0 = 0; k0 < kU; k0 += KTILE) {
#pragma unroll
    for (int i = 0; i < 16; ++i) {
      As[i * (KTILE + 1) + tid] = A[((size_t)mtile * 16 + i) * kU + k0 + tid];
    }
    __syncthreads();

#pragma unroll 8
    for (int kk = 0; kk < KTILE; kk += 4) {
      const int ksel = kk + 2 * kh;
      v2f a, b0, b1;
      a.x = As[m * (KTILE + 1) + ksel];
      a.y = As[m * (KTILE + 1) + ksel + 1];
      const size_t krow = (size_t)(k0 + ksel);
      b0.x = Bt[krow * kV + n0 + m];
      b0.y = Bt[(krow + 1) * kV + n0 + m];
      b1.x = Bt[krow * kV + n0 + 16 + m];
      b1.y = Bt[(krow + 1) * kV + n0 + 16 + m];
      acc0 = __builtin_amdgcn_wmma_f32_16x16x4_f32(false, a, false, b0,
                                                   (short)0, acc0, false, false);
      acc1 = __builtin_amdgcn_wmma_f32_16x16x4_f32(false, a, false, b1,
                                                   (short)0, acc1, false, false);
    }
    __syncthreads();
  }

#pragma unroll
  for (int j = 0; j < 8; ++j) {
    const int row = mtile * 16 + j + 8 * kh;
    const int c0  = n0 + m;
    const int c1  = n0 + 16 + m;
    out[(size_t)row * kV + c0] = acc0[j] + bias[c0];
    out[(size_t)row * kV + c1] = acc1[j] + bias[c1];
  }
}

// ---------------------------------------------------------------------------
__global__ void init_state(const float* __restrict__ c0, const float* __restrict__ h0,
                           float* __restrict__ c, float* __restrict__ h, int n) {
  const int i = blockIdx.x * blockDim.x + threadIdx.x;
  if (i < n) { c[i] = c0[i]; h[i] = h0[i]; }
}

__global__ void transpose_wout(const float* __restrict__ W,  // (256, 1024)
                               float* __restrict__ Wt) {     // (1024, 256)
  const int v = blockIdx.x;  // 0..255
  for (int k = threadIdx.x; k < kU; k += blockDim.x)
    Wt[(size_t)k * kV + v] = W[(size_t)v * kU + k];
}

// ---------------------------------------------------------------------------
extern "C" void kernel_launch(void* const* d_in, const int* in_sizes, int n_in,
                              void* d_out, int out_size, void* d_ws, size_t ws_size,
                              hipStream_t stream) {
  (void)in_sizes; (void)n_in; (void)out_size; (void)ws_size;

  const int*   tokens = (const int*)  d_in[0];   // (64, 257)
  const float* emb    = (const float*)d_in[1];   // (256, 1024)
  const float* Wl     = (const float*)d_in[2];   // (2, 2048, 4096)
  const float* gamma  = (const float*)d_in[3];   // (2, 5, 1024)
  const float* beta   = (const float*)d_in[4];   // (2, 5, 1024)
  const float* W_out  = (const float*)d_in[5];   // (256, 1024)
  const float* b_out  = (const float*)d_in[6];   // (256)
  const float* c0     = (const float*)d_in[7];   // (2, 64, 1024)
  const float* h0     = (const float*)d_in[8];   // (2, 64, 1024)
  float* out = (float*)d_out;                    // (64, 256, 256)

  char* p = (char*)d_ws;
  float* hbuf = (float*)p; p += (size_t)2 * kB * kU * sizeof(float);
  float* cbuf = (float*)p; p += (size_t)2 * kB * kU * sizeof(float);
  float* zbuf = (float*)p; p += (size_t)kB * kG * sizeof(float);
  float* Wt   = (float*)p; p += (size_t)kU * kV * sizeof(float);
  float* outs = (float*)p;                       // (64, 256, 1024) = 64 MB

  const int nstate = 2 * kB * kU;
  init_state<<<(nstate + 255) / 256, 256, 0, stream>>>(c0, h0, cbuf, hbuf, nstate);
  transpose_wout<<<kV, 256, 0, stream>>>(W_out, Wt);

  float* h0buf = hbuf;
  float* h1buf = hbuf + (size_t)kB * kU;
  float* c0buf = cbuf;
  float* c1buf = cbuf + (size_t)kB * kU;
  const float* W0 = Wl;
  const float* W1 = Wl + (size_t)kK * kG;

  for (int t = 0; t < kT; ++t) {
    // layer 0: A = [emb[tokens[:, t]] | h0]
    lstm_gemm<<<dim3(2, 32), 256, 0, stream>>>(W0, emb, h0buf, tokens + t, kT + 1, zbuf);
    lstm_pointwise<<<kB, 256, 0, stream>>>(zbuf, gamma, beta, c0buf, h0buf, nullptr, t);
    // layer 1: A = [h0_new | h1]
    lstm_gemm<<<dim3(2, 32), 256, 0, stream>>>(W1, h0buf, h1buf, nullptr, 0, zbuf);
    lstm_pointwise<<<kB, 256, 0, stream>>>(zbuf, gamma + 5 * kU, beta + 5 * kU,
                                           c1buf, h1buf, outs, t);
  }

  logits_gemm<<<kB * kT / 16, 256, 0, stream>>>(outs, Wt, b_out, out);
}